// NegSinhLinearAttention_79164837200629
// MI455X (gfx1250) — compile-verified
//
#include <hip/hip_runtime.h>
#include <hip/hip_bf16.h>

// NegSinhLinearAttention for MI455X (gfx1250), wave32 + V_WMMA_F32_16X16X4_F32.
// B=4 H=16 S=8192 D=64. Memory-bound (AI ~16 flop/byte vs 23.3 TB/s HBM).
// fp32 WMMA keeps full accuracy; the chip is nowhere near the matrix ceiling.
// sinh is computed with native v_exp_f32 (TRANS, co-issues with VALU) so the
// elementwise transcendental cost stays under the HBM streaming floor.

#define Bc 4
#define Hc 16
#define Sc 8192
#define Dc 64
#define BHc (Bc * Hc)
#define SPLIT 8
#define CHUNK (Sc / SPLIT)
#define NT 4                 // s-tiles (of 64 rows) per phase-2 block

typedef __attribute__((ext_vector_type(2))) float v2f;
typedef __attribute__((ext_vector_type(8))) float v8f;

// sinh via two native exp2 ops (v_exp_f32). Inputs are N(0,1): no overflow,
// no denorm special cases needed; absolute error ~ulp-scale.
__device__ __forceinline__ float fast_sinh(float x) {
  const float t = x * 1.4426950408889634f;   // log2(e)
  return 0.5f * (__builtin_amdgcn_exp2f(t) - __builtin_amdgcn_exp2f(-t));
}

// ---------------------------------------------------------------------------
// Phase 1: partial A_int[p][bh][d][e] = sum_{s in chunk p} Ks[s,d] * V[s,e]
//          partial kabs[p][bh][d]    = sum_{s in chunk p} |Ks[s,d]|
// Block = 128 threads (4 waves); wave w owns d-tile [16w, 16w+16).
// A fragment (16x4 f32): lanes 0-15 hold K=0,1 ; lanes 16-31 hold K=2,3.
// B fragment (4x16 f32): vgpr v, lane l -> row k = v + 2*(l>=16), col = l&15.
// ---------------------------------------------------------------------------
__global__ __launch_bounds__(128) void nsla_phase1(
    const float* __restrict__ K, const float* __restrict__ V,
    const unsigned char* __restrict__ mask,
    float* __restrict__ part_inter,   // [SPLIT][BH][Dc][Dc]
    float* __restrict__ part_kabs)    // [SPLIT][BH][Dc]
{
  const int blk = blockIdx.x;            // SPLIT * BH blocks
  const int p   = blk / BHc;
  const int bh  = blk % BHc;
  const int b   = bh / Hc;
  const int w   = threadIdx.x >> 5;      // wave id == m(d)-tile
  const int l   = threadIdx.x & 31;
  const int ln  = l & 15;
  const int hi  = l >> 4;                // half-wave selector

  const float* __restrict__ Kbh = K + (size_t)bh * Sc * Dc;
  const float* __restrict__ Vbh = V + (size_t)bh * Sc * Dc;
  const unsigned char* __restrict__ mb = mask + (size_t)b * Sc;

  const int d = 16 * w + ln;             // A-matrix row (d index) for this lane
  v8f acc[4] = {};                       // 4 n-tiles of 16x16 f32 accumulators
  float kab = 0.0f;

  const int s_begin = p * CHUNK;
  const int s_end   = s_begin + CHUNK;
  for (int s = s_begin; s < s_end; s += 4) {
    const int sa = s + 2 * hi;           // this half-wave's two K-rows
    // Streaming prefetch (global_prefetch_b8) ~8 rows ahead
    if (s + 12 < s_end) {
      __builtin_prefetch(&Kbh[(size_t)(sa + 8) * Dc + d], 0, 1);
      __builtin_prefetch(&Vbh[(size_t)(sa + 8) * Dc + ln], 0, 1);
    }
    const float m0 = mb[sa]     ? (1.0f / 64.0f) : 0.0f;
    const float m1 = mb[sa + 1] ? (1.0f / 64.0f) : 0.0f;
    v2f a;
    a.x = m0 * fast_sinh(Kbh[(size_t)sa * Dc + d]);
    a.y = m1 * fast_sinh(Kbh[(size_t)(sa + 1) * Dc + d]);
    kab += __builtin_fabsf(a.x) + __builtin_fabsf(a.y);
#pragma unroll
    for (int n = 0; n < 4; ++n) {
      const int e = 16 * n + ln;
      v2f bf;
      bf.x = Vbh[(size_t)sa * Dc + e];         // k = 2*hi
      bf.y = Vbh[(size_t)(sa + 1) * Dc + e];   // k = 2*hi + 1
      acc[n] = __builtin_amdgcn_wmma_f32_16x16x4_f32(
          false, a, false, bf, (short)0, acc[n], false, false);
    }
  }

  // D layout: vgpr v, lane l -> M = v + 8*(l>=16), N = l&15
  float* __restrict__ pi = part_inter + ((size_t)p * BHc + bh) * Dc * Dc;
#pragma unroll
  for (int n = 0; n < 4; ++n)
#pragma unroll
    for (int v = 0; v < 8; ++v) {
      const int dd = 16 * w + v + 8 * hi;
      pi[(size_t)dd * Dc + 16 * n + ln] = acc[n][v];
    }

  // |Ks| column sums: lane l and lane l+16 hold disjoint s-rows of same d
  const float tot = kab + __shfl_xor(kab, 16, 32);
  if (hi == 0)
    part_kabs[((size_t)p * BHc + bh) * Dc + 16 * w + ln] = tot;
}

// ---------------------------------------------------------------------------
// Fixed-order reduction of SPLIT partials (deterministic fp32).
// ---------------------------------------------------------------------------
__global__ __launch_bounds__(256) void nsla_reduce(
    const float* __restrict__ part_inter, const float* __restrict__ part_kabs,
    float* __restrict__ red_inter, float* __restrict__ red_kabs)
{
  const int N1 = BHc * Dc * Dc;
  const int N2 = BHc * Dc;
  const int i = blockIdx.x * 256 + threadIdx.x;
  if (i < N1) {
    float s = 0.0f;
#pragma unroll
    for (int p = 0; p < SPLIT; ++p) s += part_inter[(size_t)p * N1 + i];
    red_inter[i] = s;
  } else if (i < N1 + N2) {
    const int j = i - N1;
    float s = 0.0f;
#pragma unroll
    for (int p = 0; p < SPLIT; ++p) s += part_kabs[(size_t)p * N2 + j];
    red_kabs[j] = s;
  }
}

// ---------------------------------------------------------------------------
// Phase 2: out[s,e] = (Qs[s,:] @ A_int) / max(Qs[s,:].kabs, 0.1)
// Block = 128 threads (4 waves). One LDS fill of A_int (16 KB) + kabs is
// reused across NT=4 consecutive 64-row s-tiles (cuts L2 refill traffic 4x).
// Wave w owns rows [s0+16w, s0+16w+16) of each tile.
// ---------------------------------------------------------------------------
__global__ __launch_bounds__(128) void nsla_phase2(
    const float* __restrict__ Q,
    const float* __restrict__ red_inter, const float* __restrict__ red_kabs,
    float* __restrict__ out)
{
  __shared__ float sInt[Dc * Dc];
  __shared__ float sKab[Dc];

  const int bh  = blockIdx.x;            // 64
  const int tg  = blockIdx.y;            // 32 tile-groups of NT*64 rows
  const int tid = threadIdx.x;

  const float* __restrict__ gInt = red_inter + (size_t)bh * Dc * Dc;
  for (int i = tid; i < Dc * Dc; i += 128) sInt[i] = gInt[i];
  if (tid < Dc) sKab[tid] = red_kabs[(size_t)bh * Dc + tid];
  __syncthreads();

  const int w  = tid >> 5;
  const int l  = tid & 31;
  const int ln = l & 15;
  const int hi = l >> 4;

  float* __restrict__ obh = out + (size_t)bh * Sc * Dc;

  for (int t = 0; t < NT; ++t) {
    const int s0  = (tg * NT + t) * 64 + 16 * w;
    const int row = s0 + ln;             // this lane's A row
    const float* __restrict__ Qrow = Q + ((size_t)bh * Sc + row) * Dc;

    // Load + sinh the 64-wide Q row (split across the two half-waves),
    // fused with the norm dot product against kabs.
    v2f qs[16];
    float nrm = 0.0f;
#pragma unroll
    for (int kk = 0; kk < 16; ++kk) {
      const int c0 = 4 * kk + 2 * hi;
      const v2f q2 = *(const v2f*)(Qrow + c0);
      qs[kk].x = fast_sinh(q2.x);
      qs[kk].y = fast_sinh(q2.y);
      nrm += qs[kk].x * sKab[c0] + qs[kk].y * sKab[c0 + 1];
    }
    nrm += __shfl_xor(nrm, 16, 32);      // combine the two half-rows
    const float rn = 1.0f / fmaxf(nrm, 0.1f);

    v8f acc[4] = {};
#pragma unroll
    for (int kk = 0; kk < 16; ++kk) {
      const v2f a = qs[kk];
#pragma unroll
      for (int n = 0; n < 4; ++n) {
        v2f bf;
        bf.x = sInt[(4 * kk + 2 * hi) * Dc + 16 * n + ln];
        bf.y = sInt[(4 * kk + 2 * hi + 1) * Dc + 16 * n + ln];
        acc[n] = __builtin_amdgcn_wmma_f32_16x16x4_f32(
            false, a, false, bf, (short)0, acc[n], false, false);
      }
    }

    // Scale by 1/norm of the row each D-element belongs to, then store
    // (64 B coalesced row segments across lanes 0-15 / 16-31).
#pragma unroll
    for (int v = 0; v < 8; ++v) {
      const float rnv = __shfl(rn, v + 8 * hi, 32);  // norm lives in lane == row-in-tile
      const int r = s0 + v + 8 * hi;
#pragma unroll
      for (int n = 0; n < 4; ++n)
        obh[(size_t)r * Dc + 16 * n + ln] = acc[n][v] * rnv;
    }
  }
}

// ---------------------------------------------------------------------------
extern "C" void kernel_launch(void* const* d_in, const int* in_sizes, int n_in,
                              void* d_out, int out_size, void* d_ws, size_t ws_size,
                              hipStream_t stream) {
  const float* Q = (const float*)d_in[0];
  const float* K = (const float*)d_in[1];
  const float* V = (const float*)d_in[2];
  const unsigned char* mask = (const unsigned char*)d_in[3];
  float* out = (float*)d_out;

  // Workspace layout (floats): partials then reduced tensors (~9.3 MB total)
  float* part_inter = (float*)d_ws;                                  // SPLIT*BH*64*64
  float* part_kabs  = part_inter + (size_t)SPLIT * BHc * Dc * Dc;    // SPLIT*BH*64
  float* red_inter  = part_kabs + (size_t)SPLIT * BHc * Dc;          // BH*64*64
  float* red_kabs   = red_inter + (size_t)BHc * Dc * Dc;             // BH*64

  nsla_phase1<<<dim3(SPLIT * BHc), 128, 0, stream>>>(K, V, mask, part_inter, part_kabs);

  const int total = BHc * Dc * Dc + BHc * Dc;
  nsla_reduce<<<dim3((total + 255) / 256), 256, 0, stream>>>(part_inter, part_kabs,
                                                             red_inter, red_kabs);

  nsla_phase2<<<dim3(BHc, Sc / (64 * NT)), 128, 0, stream>>>(Q, red_inter, red_kabs, out);
}